// IHRLayer_30468497997973
// MI455X (gfx1250) — compile-verified
//
#include <hip/hip_runtime.h>
#include <math.h>

// ---------------------------------------------------------------------------
// IHR forward for MI455X (gfx1250, wave32, WMMA).
// Pipeline:
//   K0 prep      : q=bq+bp, off/attn reductions, z-chain, camera projections
//   K1 sample    : bilinear gather from feat, attn-weighted sum  -> X (Q x C)
//   K2 gemm      : Y = X @ w_out + b_out + bev_q, store transposed (C x Q)
//                  [v_wmma_f32_16x16x4_f32]
//   K3 conv+bn   : 3x3 conv as implicit GEMM over 9 shifted taps + BN + ReLU
//                  [v_wmma_f32_16x16x4_f32]
//   K4 chmean    : per-channel spatial mean
//   K5 se        : s = sigmoid(mean @ se_w + se_b)
//   K6 se-apply  : x *= s ; channel max / mean maps for CBAM
//   K7 cbam      : 7x7 conv (2->1) + sigmoid -> spatial attention
//   K8 final     : out[q,c] = x[c,q] * a[q]  (transpose to (Q,C))
// ---------------------------------------------------------------------------

#define QN    16384      // queries / pixels (128*128)
#define CN    256        // channels
#define NCAM  6
#define FH    32
#define FW    88
#define FHW   (FH*FW)    // 2816
#define BHW   128
#define EPSF  1e-5f

typedef float v2f __attribute__((ext_vector_type(2)));
typedef float v8f __attribute__((ext_vector_type(8)));

__device__ __forceinline__ float sigmoidf(float x) {
    return 1.0f / (1.0f + __expf(-x));
}

// ---------------------------------------------------------------------------
// K0: per-query prep. One block per query, 256 threads (one per channel).
// ---------------------------------------------------------------------------
__global__ void prep_kernel(const float* __restrict__ bev_q,
                            const float* __restrict__ bev_pos,
                            const float* __restrict__ refp,
                            const float* __restrict__ l2i,
                            const float* __restrict__ aug,
                            const float* __restrict__ w_pos,
                            const float* __restrict__ b_pos,
                            const float* __restrict__ w_attn,
                            const float* __restrict__ b_attn,
                            float* __restrict__ proj,     // Q * 18 : (ix,iy,w)*6
                            float* __restrict__ rp_out)   // Q * 3
{
    __shared__ float red[CN];
    __shared__ float sums[7];
    const int q = blockIdx.x;
    const int c = threadIdx.x;

    const float qc = bev_q[q*CN + c] + bev_pos[q*CN + c];
    float part[7];
    part[0] = qc * w_pos[c];
#pragma unroll
    for (int n = 0; n < NCAM; ++n) part[1+n] = qc * w_attn[c*NCAM + n];

    for (int s = 0; s < 7; ++s) {
        red[c] = part[s];
        __syncthreads();
        for (int off = 128; off > 0; off >>= 1) {
            if (c < off) red[c] += red[c + off];
            __syncthreads();
        }
        if (c == 0) sums[s] = red[0];
        __syncthreads();
    }

    if (c == 0) {
        const float off = sums[0] + b_pos[0];
        float z0 = refp[q*3 + 2];
        z0 = fminf(fmaxf(z0, 0.0f), 1.0f);
        const float zl = __logf(fmaxf(z0, EPSF) / fmaxf(1.0f - z0, EPSF));
        const float z  = sigmoidf(zl + off);
        const float rx = refp[q*3 + 0];
        const float ry = refp[q*3 + 1];
        rp_out[q*3 + 0] = rx;
        rp_out[q*3 + 1] = ry;
        rp_out[q*3 + 2] = z;

        const float px = rx * 102.4f - 51.2f;
        const float py = ry * 102.4f - 51.2f;
        const float pz = z  * 8.0f   - 5.0f;

#pragma unroll
        for (int n = 0; n < NCAM; ++n) {
            const float* L = l2i + n*12;
            const float c0 = L[0]*px + L[1]*py + L[2]*pz  + L[3];
            const float c1 = L[4]*px + L[5]*py + L[6]*pz  + L[7];
            const float c2 = L[8]*px + L[9]*py + L[10]*pz + L[11];
            bool m = c2 > EPSF;
            const float d  = fmaxf(c2, EPSF);
            const float f0 = c0 / d, f1 = c1 / d;
            const float* A = aug + n*12;
            float u = A[0]*f0 + A[1]*f1 + A[2]*c2 + A[3];
            float v = A[4]*f0 + A[5]*f1 + A[6]*c2 + A[7];
            u = (u / 704.0f - 0.5f) * 2.0f;
            v = (v / 256.0f - 0.5f) * 2.0f;
            m = m && (u > -1.0f) && (u < 1.0f) && (v > -1.0f) && (v < 1.0f);
            const float attn = sigmoidf(sums[1+n] + b_attn[n]);
            const float w = m ? attn : 0.0f;
            // grid-sample pixel coords on the 32x88 feature map
            const float ix = (u + 1.0f) * (FW * 0.5f) - 0.5f;
            const float iy = (v + 1.0f) * (FH * 0.5f) - 0.5f;
            proj[q*18 + n*3 + 0] = ix;
            proj[q*18 + n*3 + 1] = iy;
            proj[q*18 + n*3 + 2] = w;
        }
    }
}

// ---------------------------------------------------------------------------
// K1: bilinear sample + attn-weighted camera sum -> X[q][c]
// One block per query, thread = channel. Corner gathers hit L2 (feat=17.3MB).
// ---------------------------------------------------------------------------
__global__ void sample_kernel(const float* __restrict__ feat,
                              const float* __restrict__ proj,
                              float* __restrict__ X)
{
    const int q = blockIdx.x;
    const int c = threadIdx.x;
    float acc = 0.0f;
#pragma unroll
    for (int n = 0; n < NCAM; ++n) {
        const float w = proj[q*18 + n*3 + 2];
        if (w == 0.0f) continue;
        const float ix = proj[q*18 + n*3 + 0];
        const float iy = proj[q*18 + n*3 + 1];
        const float x0 = floorf(ix), y0 = floorf(iy);
        const float wx1 = ix - x0, wy1 = iy - y0;
        const float wx0 = 1.0f - wx1, wy0 = 1.0f - wy1;
        const int xi0 = (int)x0, yi0 = (int)y0;
        const int xi1 = xi0 + 1, yi1 = yi0 + 1;
        const float* f = feat + (n*CN + c) * FHW;
        float s = 0.0f;
        if (xi0 >= 0 && xi0 < FW && yi0 >= 0 && yi0 < FH) s += wx0*wy0*f[yi0*FW + xi0];
        if (xi1 >= 0 && xi1 < FW && yi0 >= 0 && yi0 < FH) s += wx1*wy0*f[yi0*FW + xi1];
        if (xi0 >= 0 && xi0 < FW && yi1 >= 0 && yi1 < FH) s += wx0*wy1*f[yi1*FW + xi0];
        if (xi1 >= 0 && xi1 < FW && yi1 >= 0 && yi1 < FH) s += wx1*wy1*f[yi1*FW + xi1];
        acc += w * s;
    }
    X[q*CN + c] = acc;
}

// ---------------------------------------------------------------------------
// K2: Y = X(QxC) @ W(CxC) + b_out + bev_q, stored transposed as convin[c][q].
// One 16x16 output tile per wave; K stepped by 4 via v_wmma_f32_16x16x4_f32.
// A layout (16x4 f32): lane holds M=lane&15; K-offset 0 (lanes 0-15) / 2
// (lanes 16-31) across the 2 VGPRs.  B (4x16) mirrored with N=lane&15.
// ---------------------------------------------------------------------------
__global__ void gemm_out_kernel(const float* __restrict__ X,
                                const float* __restrict__ W,
                                const float* __restrict__ bias,
                                const float* __restrict__ bevq,
                                float* __restrict__ convin)
{
    const int wave = threadIdx.x >> 5;
    const int lane = threadIdx.x & 31;
    const int tile = blockIdx.x * 8 + wave;     // 16384 tiles
    const int tn = (tile & 15) * 16;            // channel (N) block
    const int tm = (tile >> 4) * 16;            // query  (M) block
    const int mrow = lane & 15;
    const int koff = (lane < 16) ? 0 : 2;

    v8f acc = {};
    for (int k0 = 0; k0 < CN; k0 += 4) {
        const float* ap = X + (tm + mrow)*CN + k0 + koff;
        v2f a; a.x = ap[0]; a.y = ap[1];
        const float* bp = W + (k0 + koff)*CN + tn + mrow;
        v2f b; b.x = bp[0]; b.y = bp[CN];
        acc = __builtin_amdgcn_wmma_f32_16x16x4_f32(
            false, a, false, b, (short)0, acc, false, false);
    }

    const int ng    = tn + mrow;
    const int mbase = tm + ((lane < 16) ? 0 : 8);
#pragma unroll
    for (int r = 0; r < 8; ++r) {
        const int qg = mbase + r;
        const float v = acc[r] + bias[ng] + bevq[qg*CN + ng];
        convin[ng*QN + qg] = v;   // transposed store -> (C, 128, 128) layout
    }
}

// ---------------------------------------------------------------------------
// K3: 3x3 conv (256->256) as implicit GEMM over 9 shifted taps, + BN + ReLU.
// Tile = 16 output channels x 16 consecutive pixels (one image row slice).
// y-tap boundary is wave-uniform (scalar branch, EXEC stays all-1s);
// x boundary handled with per-lane selects feeding zeros into B.
// ---------------------------------------------------------------------------
__global__ void conv_bn_kernel(const float* __restrict__ convin,
                               const float* __restrict__ cw,   // (256,256,3,3)
                               const float* __restrict__ cb,
                               const float* __restrict__ gamma,
                               const float* __restrict__ beta,
                               const float* __restrict__ mean,
                               const float* __restrict__ var,
                               float* __restrict__ xbn)
{
    const int wave = threadIdx.x >> 5;
    const int lane = threadIdx.x & 31;
    const int tile = blockIdx.x * 8 + wave;     // 16 * 1024 tiles
    const int tn = (tile & 1023) * 16;          // pixel block base
    const int tm = (tile >> 10) * 16;           // cout block base
    const int y  = tn >> 7;                     // all 16 pixels share row y
    const int xb = tn & 127;
    const int nlane = lane & 15;
    const int koff  = (lane < 16) ? 0 : 2;
    const int co    = tm + nlane;

    v8f acc = {};
    for (int t = 0; t < 9; ++t) {
        const int dy = t / 3 - 1, dx = t % 3 - 1;
        const int yy = y + dy;
        if (yy < 0 || yy >= BHW) continue;      // uniform across wave
        const int xx = xb + nlane + dx;
        const bool valid = (xx >= 0) && (xx < BHW);
        const int psh = yy * BHW + xx;
        for (int k0 = 0; k0 < CN; k0 += 4) {
            const int ci = k0 + koff;
            v2f a;
            a.x = cw[(co*CN + ci    )*9 + t];
            a.y = cw[(co*CN + ci + 1)*9 + t];
            v2f b;
            b.x = valid ? convin[(ci    )*QN + psh] : 0.0f;
            b.y = valid ? convin[(ci + 1)*QN + psh] : 0.0f;
            acc = __builtin_amdgcn_wmma_f32_16x16x4_f32(
                false, a, false, b, (short)0, acc, false, false);
        }
    }

    const int mbase = tm + ((lane < 16) ? 0 : 8);
    const int pg    = tn + nlane;
#pragma unroll
    for (int r = 0; r < 8; ++r) {
        const int cg = mbase + r;
        float v = acc[r] + cb[cg];
        v = gamma[cg] * (v - mean[cg]) * rsqrtf(var[cg] + 1e-5f) + beta[cg];
        v = fmaxf(v, 0.0f);                     // ReLU
        xbn[cg*QN + pg] = v;
    }
}

// ---------------------------------------------------------------------------
// K4: per-channel spatial mean (one block per channel)
// ---------------------------------------------------------------------------
__global__ void chmean_kernel(const float* __restrict__ xbn,
                              float* __restrict__ chmean)
{
    __shared__ float red[256];
    const int c = blockIdx.x, t = threadIdx.x;
    float s = 0.0f;
    for (int p = t; p < QN; p += 256) s += xbn[c*QN + p];
    red[t] = s; __syncthreads();
    for (int off = 128; off > 0; off >>= 1) {
        if (t < off) red[t] += red[t + off];
        __syncthreads();
    }
    if (t == 0) chmean[c] = red[0] * (1.0f / (float)QN);
}

// ---------------------------------------------------------------------------
// K5: SE gate, single block GEMV 256x256
// ---------------------------------------------------------------------------
__global__ void se_kernel(const float* __restrict__ chmean,
                          const float* __restrict__ se_w,
                          const float* __restrict__ se_b,
                          float* __restrict__ svec)
{
    __shared__ float m[256];
    const int j = threadIdx.x;
    m[j] = chmean[j];
    __syncthreads();
    float s = se_b[j];
    for (int c = 0; c < CN; ++c) s += m[c] * se_w[c*CN + j];
    svec[j] = sigmoidf(s);
}

// ---------------------------------------------------------------------------
// K6: apply SE scale; compute CBAM channel max & mean per pixel
// ---------------------------------------------------------------------------
__global__ void se_apply_kernel(const float* __restrict__ xbn,
                                const float* __restrict__ svec,
                                float* __restrict__ xse,
                                float* __restrict__ mxb,
                                float* __restrict__ avb)
{
    const int p = blockIdx.x * 256 + threadIdx.x;
    float mx = -3.4e38f, sum = 0.0f;
    for (int c = 0; c < CN; ++c) {
        const float v = xbn[c*QN + p] * svec[c];
        xse[c*QN + p] = v;
        mx = fmaxf(mx, v);
        sum += v;
    }
    mxb[p] = mx;
    avb[p] = sum * (1.0f / (float)CN);
}

// ---------------------------------------------------------------------------
// K7: CBAM spatial attention: 7x7 conv over (max,mean), pad 3, sigmoid
// ---------------------------------------------------------------------------
__global__ void cbam_kernel(const float* __restrict__ mxb,
                            const float* __restrict__ avb,
                            const float* __restrict__ cw,  // (1,2,7,7)
                            const float* __restrict__ cb,
                            float* __restrict__ avec)
{
    const int p = blockIdx.x * 256 + threadIdx.x;
    const int y = p >> 7, x = p & 127;
    float acc = cb[0];
    for (int ky = 0; ky < 7; ++ky) {
        const int yy = y + ky - 3;
        if (yy < 0 || yy >= BHW) continue;
        for (int kx = 0; kx < 7; ++kx) {
            const int xx = x + kx - 3;
            if (xx < 0 || xx >= BHW) continue;
            const int pp = yy * BHW + xx;
            acc += cw[ky*7 + kx] * mxb[pp] + cw[49 + ky*7 + kx] * avb[pp];
        }
    }
    avec[p] = sigmoidf(acc);
}

// ---------------------------------------------------------------------------
// K8: final transpose + spatial-attention multiply -> out (Q, C)
// ---------------------------------------------------------------------------
__global__ void final_kernel(const float* __restrict__ xse,
                             const float* __restrict__ avec,
                             float* __restrict__ out)
{
    const int q = blockIdx.x;
    const int c = threadIdx.x;
    out[q*CN + c] = xse[c*QN + q] * avec[q];
}

// ---------------------------------------------------------------------------
extern "C" void kernel_launch(void* const* d_in, const int* in_sizes, int n_in,
                              void* d_out, int out_size, void* d_ws, size_t ws_size,
                              hipStream_t stream)
{
    (void)in_sizes; (void)n_in; (void)out_size; (void)ws_size;
    const float* feat     = (const float*)d_in[0];
    const float* bev_q    = (const float*)d_in[1];
    const float* bev_pos  = (const float*)d_in[2];
    const float* refp     = (const float*)d_in[3];
    const float* l2i      = (const float*)d_in[4];
    const float* aug      = (const float*)d_in[5];
    const float* w_pos    = (const float*)d_in[6];
    const float* b_pos    = (const float*)d_in[7];
    const float* w_attn   = (const float*)d_in[8];
    const float* b_attn   = (const float*)d_in[9];
    const float* w_out    = (const float*)d_in[10];
    const float* b_out    = (const float*)d_in[11];
    const float* conv_w   = (const float*)d_in[12];
    const float* conv_b   = (const float*)d_in[13];
    const float* bn_gamma = (const float*)d_in[14];
    const float* bn_beta  = (const float*)d_in[15];
    const float* bn_mean  = (const float*)d_in[16];
    const float* bn_var   = (const float*)d_in[17];
    const float* se_w     = (const float*)d_in[18];
    const float* se_b     = (const float*)d_in[19];
    const float* cbam_w   = (const float*)d_in[20];
    const float* cbam_b   = (const float*)d_in[21];

    float* out_main = (float*)d_out;                  // (1, Q, C)
    float* rp_out   = (float*)d_out + (size_t)QN*CN;  // (1, 1, Q, 3)

    // workspace layout (floats)
    float* ws = (float*)d_ws;
    float* bufA   = ws;                          // X (Q,C) then xbn (C,Q)
    float* bufB   = ws + (size_t)QN*CN;          // convin (C,Q) then xse (C,Q)
    float* proj   = ws + (size_t)2*QN*CN;        // Q*18
    float* chmean = proj + (size_t)QN*18;        // 256
    float* svec   = chmean + 256;                // 256
    float* mxb    = svec + 256;                  // Q
    float* avb    = mxb + QN;                    // Q
    float* avec   = avb + QN;                    // Q

    // K0: prep (also writes rp into d_out)
    prep_kernel<<<QN, 256, 0, stream>>>(bev_q, bev_pos, refp, l2i, aug,
                                        w_pos, b_pos, w_attn, b_attn,
                                        proj, rp_out);
    // K1: bilinear sample -> X in bufA
    sample_kernel<<<QN, 256, 0, stream>>>(feat, proj, bufA);
    // K2: output projection GEMM (WMMA) -> convin in bufB (transposed)
    gemm_out_kernel<<<(QN/16)*(CN/16)/8, 256, 0, stream>>>(bufA, w_out, b_out,
                                                           bev_q, bufB);
    // K3: 3x3 conv implicit GEMM (WMMA) + BN + ReLU -> xbn in bufA
    conv_bn_kernel<<<(CN/16)*(QN/16)/8, 256, 0, stream>>>(bufB, conv_w, conv_b,
                                                          bn_gamma, bn_beta,
                                                          bn_mean, bn_var, bufA);
    // K4: channel means
    chmean_kernel<<<CN, 256, 0, stream>>>(bufA, chmean);
    // K5: SE gate
    se_kernel<<<1, 256, 0, stream>>>(chmean, se_w, se_b, svec);
    // K6: apply SE; CBAM stats -> xse in bufB
    se_apply_kernel<<<QN/256, 256, 0, stream>>>(bufA, svec, bufB, mxb, avb);
    // K7: CBAM 7x7 conv + sigmoid
    cbam_kernel<<<QN/256, 256, 0, stream>>>(mxb, avb, cbam_w, cbam_b, avec);
    // K8: final transpose + multiply -> d_out main
    final_kernel<<<QN, 256, 0, stream>>>(bufB, avec, out_main);
}